// GAT_20933670600831
// MI455X (gfx1250) — compile-verified
//
#include <hip/hip_runtime.h>
#include <hip/hip_bf16.h>

typedef float v2f __attribute__((ext_vector_type(2)));
typedef float v8f __attribute__((ext_vector_type(8)));

// ---------- helpers: monotonic uint encoding of float for atomic max ----------
__device__ __forceinline__ unsigned fenc(float f) {
    unsigned u = __float_as_uint(f);
    return (u & 0x80000000u) ? ~u : (u | 0x80000000u);
}
__device__ __forceinline__ float fdec2(unsigned u) {
    unsigned v = (u & 0x80000000u) ? (u & 0x7FFFFFFFu) : ~u;
    return __uint_as_float(v);
}

// ---------- zero fill ----------
__global__ void fill_zero_u32(unsigned* __restrict__ p, int n) {
    int i = blockIdx.x * blockDim.x + threadIdx.x;
    if (i < n) p[i] = 0u;
}

// ---------- edge_attr column means (deterministic two-stage reduction) ----------
__global__ void ea_partial_kernel(const float* __restrict__ ea, float* __restrict__ part, int E) {
    __shared__ float s0[256], s1[256];
    float a0 = 0.f, a1 = 0.f;
    for (int e = blockIdx.x * blockDim.x + threadIdx.x; e < E; e += gridDim.x * blockDim.x) {
        a0 += ea[2 * e];
        a1 += ea[2 * e + 1];
    }
    s0[threadIdx.x] = a0; s1[threadIdx.x] = a1;
    __syncthreads();
    for (int s = 128; s > 0; s >>= 1) {
        if ((int)threadIdx.x < s) {
            s0[threadIdx.x] += s0[threadIdx.x + s];
            s1[threadIdx.x] += s1[threadIdx.x + s];
        }
        __syncthreads();
    }
    if (threadIdx.x == 0) { part[2 * blockIdx.x] = s0[0]; part[2 * blockIdx.x + 1] = s1[0]; }
}

__global__ void ea_final_kernel(const float* __restrict__ part, float* __restrict__ mean,
                                int nblocks, int E) {
    __shared__ float s0[256], s1[256];
    float a0 = 0.f, a1 = 0.f;
    for (int i = threadIdx.x; i < nblocks; i += 256) { a0 += part[2 * i]; a1 += part[2 * i + 1]; }
    s0[threadIdx.x] = a0; s1[threadIdx.x] = a1;
    __syncthreads();
    for (int s = 128; s > 0; s >>= 1) {
        if ((int)threadIdx.x < s) {
            s0[threadIdx.x] += s0[threadIdx.x + s];
            s1[threadIdx.x] += s1[threadIdx.x + s];
        }
        __syncthreads();
    }
    if (threadIdx.x == 0) { mean[0] = s0[0] / (float)E; mean[1] = s1[0] / (float)E; }
}

// ---------- fp32 WMMA GEMM: Y[M x N] = X[M x K] @ W[K x N] + bias ----------
// block = 128 threads (4 waves); block tile = 16 rows x 64 cols; grid = (M/16, N/64)
// K, N compile-time so every W/Y access folds into a load/store immediate offset.
template <int K, int N>
__global__ void gemm_wmma_bias(const float* __restrict__ X, const float* __restrict__ W,
                               const float* __restrict__ bias, float* __restrict__ Y, int M) {
    constexpr int KP = K + 4;  // LDS pad: bank-conflict-free, keeps 16B alignment
    __shared__ float lds[16 * KP];
    const int lane = threadIdx.x & 31;
    const int wave = threadIdx.x >> 5;
    const int row0 = blockIdx.x * 16;

    // cooperative stage of X tile [16 x K] into LDS (coalesced along K)
    for (int idx = threadIdx.x; idx < 16 * K; idx += blockDim.x) {
        int r = idx / K, c = idx - r * K;
        int m = row0 + r;
        lds[r * KP + c] = (m < M) ? X[m * K + c] : 0.0f;
    }
    __syncthreads();

    const int half = lane >> 4;               // 0: lanes 0-15, 1: lanes 16-31
    const int nn = blockIdx.y * 64 + wave * 16 + (lane & 15);
    const int bk = half << 1;                 // which K-pair this half-wave holds
    const int mrow = (lane & 15);             // A-frag row for this lane

    const float* Wp = W + bk * N + nn;        // per-lane base; loop offsets are constants

    v8f acc = {};
#pragma unroll
    for (int kk = 0; kk < K; kk += 4) {
        v2f a, b;
        a.x = lds[mrow * KP + kk + bk];
        a.y = lds[mrow * KP + kk + bk + 1];
        b.x = Wp[kk * N];                     // compile-time immediate offsets
        b.y = Wp[kk * N + N];
        acc = __builtin_amdgcn_wmma_f32_16x16x4_f32(false, a, false, b, (short)0, acc,
                                                    false, false);
    }

    const float bv = bias[nn];
    const int mbase = row0 + (half << 3);     // VGPR r holds M = mbase + r
    float* Yp = Y + mbase * N + nn;
#pragma unroll
    for (int r = 0; r < 8; ++r) {
        if (mbase + r < M) Yp[r * N] = acc[r] + bv;
    }
}

// ---------- edge pass A: attention logits + segment max ----------
template <int H, int C>
__global__ void edge_logits_kernel(const float* __restrict__ xl, const float* __restrict__ xr,
                                   const int* __restrict__ ei, const float* __restrict__ ea,
                                   const float* __restrict__ eamean,
                                   const float* __restrict__ We, const float* __restrict__ be,
                                   const float* __restrict__ att,
                                   float* __restrict__ logits, unsigned* __restrict__ lmax,
                                   int Ebase, int Etot) {
    int t = blockIdx.x * blockDim.x + threadIdx.x;
    if (t >= Etot * H) return;
    int e = t / H, h = t - e * H;
    int s, d;
    float a0, a1;
    if (e < Ebase) {
        s = ei[e];
        d = ei[Ebase + e];
        a0 = ea[2 * e];
        a1 = ea[2 * e + 1];
    } else {
        s = d = e - Ebase;
        a0 = eamean[0];
        a1 = eamean[1];
    }
    const float4* xls = (const float4*)(xl + s * (H * C) + h * C);
    const float4* xrd = (const float4*)(xr + d * (H * C) + h * C);
    const float4* W0 = (const float4*)(We + h * C);
    const float4* W1 = (const float4*)(We + H * C + h * C);
    const float4* BE = (const float4*)(be + h * C);
    const float4* AT = (const float4*)(att + h * C);

    float acc = 0.f;
#pragma unroll
    for (int c = 0; c < C / 4; ++c) {
        float4 xa = xls[c], xb = xrd[c], w0 = W0[c], w1 = W1[c], bb = BE[c], at = AT[c];
        float m;
        m = xa.x + xb.x + a0 * w0.x + a1 * w1.x + bb.x; m = m > 0.f ? m : 0.2f * m; acc += at.x * m;
        m = xa.y + xb.y + a0 * w0.y + a1 * w1.y + bb.y; m = m > 0.f ? m : 0.2f * m; acc += at.y * m;
        m = xa.z + xb.z + a0 * w0.z + a1 * w1.z + bb.z; m = m > 0.f ? m : 0.2f * m; acc += at.z * m;
        m = xa.w + xb.w + a0 * w0.w + a1 * w1.w + bb.w; m = m > 0.f ? m : 0.2f * m; acc += at.w * m;
    }
    logits[t] = acc;
    atomicMax(lmax + d * H + h, fenc(acc));
}

// ---------- edge pass B: exp(logit - max), accumulate denominator ----------
template <int H>
__global__ void edge_softmax_kernel(const int* __restrict__ ei, float* __restrict__ logits,
                                    const unsigned* __restrict__ lmax, float* __restrict__ denom,
                                    int Ebase, int Etot) {
    int t = blockIdx.x * blockDim.x + threadIdx.x;
    if (t >= Etot * H) return;
    int e = t / H, h = t - e * H;
    int d = (e < Ebase) ? ei[Ebase + e] : (e - Ebase);
    float lm = fdec2(lmax[d * H + h]);
    float ex = expf(logits[t] - lm);
    logits[t] = ex;  // overwrite: pass C reads exp values
    atomicAdd(denom + d * H + h, ex);
}

// ---------- edge pass C: alpha-weighted scatter aggregate ----------
template <int H, int C>
__global__ void edge_aggregate_kernel(const float* __restrict__ xl, const int* __restrict__ ei,
                                      const float* __restrict__ logits,
                                      const float* __restrict__ denom, float* __restrict__ out,
                                      int Ebase, int Etot) {
    int t = blockIdx.x * blockDim.x + threadIdx.x;
    if (t >= Etot * H) return;
    int e = t / H, h = t - e * H;
    int s, d;
    if (e < Ebase) { s = ei[e]; d = ei[Ebase + e]; }
    else           { s = d = e - Ebase; }
    float alpha = logits[t] / denom[d * H + h];
    const float* xs = xl + s * (H * C) + h * C;
    float* o = out + d * (H * C) + h * C;
#pragma unroll 8
    for (int c = 0; c < C; ++c) atomicAdd(o + c, xs[c] * alpha);
}

// ---------- finalize: add bias (optionally ReLU); ncols must be power of two ----------
__global__ void bias_act_kernel(float* __restrict__ y, const float* __restrict__ bias,
                                int n, int colmask, int do_relu) {
    int i = blockIdx.x * blockDim.x + threadIdx.x;
    if (i >= n) return;
    float v = y[i] + bias[i & colmask];
    y[i] = do_relu ? fmaxf(v, 0.f) : v;
}

// =====================================================================
extern "C" void kernel_launch(void* const* d_in, const int* in_sizes, int n_in,
                              void* d_out, int out_size, void* d_ws, size_t ws_size,
                              hipStream_t stream) {
    (void)n_in; (void)out_size; (void)ws_size;
    const float* x     = (const float*)d_in[0];
    const int*   ei    = (const int*)d_in[1];
    const float* ea    = (const float*)d_in[2];
    const float* W1l   = (const float*)d_in[3];
    const float* b1l   = (const float*)d_in[4];
    const float* W1r   = (const float*)d_in[5];
    const float* b1r   = (const float*)d_in[6];
    const float* We1   = (const float*)d_in[7];
    const float* be1   = (const float*)d_in[8];
    const float* att1  = (const float*)d_in[9];
    const float* bias1 = (const float*)d_in[10];
    const float* W2l   = (const float*)d_in[11];
    const float* b2l   = (const float*)d_in[12];
    const float* W2r   = (const float*)d_in[13];
    const float* b2r   = (const float*)d_in[14];
    const float* We2   = (const float*)d_in[15];
    const float* be2   = (const float*)d_in[16];
    const float* att2  = (const float*)d_in[17];
    const float* bias2 = (const float*)d_in[18];

    const int N    = in_sizes[0] / 128;  // nodes
    const int E    = in_sizes[1] / 2;    // original edges
    const int Etot = E + N;              // + self loops

    // ---- workspace carve-up (256B aligned) ----
    char* wsb = (char*)d_ws;
    size_t off = 0;
    auto wsalloc = [&](size_t bytes) -> void* {
        void* p = wsb + off;
        off += (bytes + 255) & ~(size_t)255;
        return p;
    };
    float*    xl1    = (float*)wsalloc((size_t)N * 128 * 4);
    float*    xr1    = (float*)wsalloc((size_t)N * 128 * 4);
    float*    acc1   = (float*)wsalloc((size_t)N * 128 * 4);  // layer1 aggregate -> h
    float*    logits = (float*)wsalloc((size_t)Etot * 4 * 4); // reused by layer2 (needs Etot)
    unsigned* lmax1  = (unsigned*)wsalloc((size_t)N * 4 * 4);
    float*    den1   = (float*)wsalloc((size_t)N * 4 * 4);
    unsigned* lmax2  = (unsigned*)wsalloc((size_t)N * 4);
    float*    den2   = (float*)wsalloc((size_t)N * 4);
    float*    part   = (float*)wsalloc(256 * 2 * 4);
    float*    eamean = (float*)wsalloc(2 * 4);
    float*    xl2 = xl1;  // layer1 xl dead after layer1 pass C -> alias
    float*    xr2 = xr1;
    float*    out = (float*)d_out;

    auto blks = [](int n) { return dim3((unsigned)((n + 255) / 256)); };

    // ---- zero all accumulators (required every call: graph replay + poisoned ws) ----
    fill_zero_u32<<<blks(N * 4), 256, 0, stream>>>(lmax1, N * 4);
    fill_zero_u32<<<blks(N * 4), 256, 0, stream>>>((unsigned*)den1, N * 4);
    fill_zero_u32<<<blks(N * 128), 256, 0, stream>>>((unsigned*)acc1, N * 128);
    fill_zero_u32<<<blks(N), 256, 0, stream>>>(lmax2, N);
    fill_zero_u32<<<blks(N), 256, 0, stream>>>((unsigned*)den2, N);
    fill_zero_u32<<<blks(N * 64), 256, 0, stream>>>((unsigned*)out, N * 64);

    // ---- self-loop edge_attr = column mean of edge_attr ----
    ea_partial_kernel<<<256, 256, 0, stream>>>(ea, part, E);
    ea_final_kernel<<<1, 256, 0, stream>>>(part, eamean, 256, E);

    // ---- layer 1 dense transforms (fp32 WMMA) ----
    dim3 gblk(128);
    dim3 gg1((N + 15) / 16, 2);   // 2 x 64 cols = 128
    gemm_wmma_bias<128, 128><<<gg1, gblk, 0, stream>>>(x, W1l, b1l, xl1, N);
    gemm_wmma_bias<128, 128><<<gg1, gblk, 0, stream>>>(x, W1r, b1r, xr1, N);

    // ---- layer 1 edge phase (H=4, C=32) ----
    const int T1 = Etot * 4;
    edge_logits_kernel<4, 32><<<blks(T1), 256, 0, stream>>>(xl1, xr1, ei, ea, eamean, We1, be1,
                                                            att1, logits, lmax1, E, Etot);
    edge_softmax_kernel<4><<<blks(T1), 256, 0, stream>>>(ei, logits, lmax1, den1, E, Etot);
    edge_aggregate_kernel<4, 32><<<blks(T1), 256, 0, stream>>>(xl1, ei, logits, den1, acc1, E, Etot);
    bias_act_kernel<<<blks(N * 128), 256, 0, stream>>>(acc1, bias1, N * 128, 127, 1);

    // ---- layer 2 dense transforms (input = acc1 == h) ----
    dim3 gg2((N + 15) / 16, 1);
    gemm_wmma_bias<128, 64><<<gg2, gblk, 0, stream>>>(acc1, W2l, b2l, xl2, N);
    gemm_wmma_bias<128, 64><<<gg2, gblk, 0, stream>>>(acc1, W2r, b2r, xr2, N);

    // ---- layer 2 edge phase (H=1, C=64) ----
    const int T2 = Etot;
    edge_logits_kernel<1, 64><<<blks(T2), 256, 0, stream>>>(xl2, xr2, ei, ea, eamean, We2, be2,
                                                            att2, logits, lmax2, E, Etot);
    edge_softmax_kernel<1><<<blks(T2), 256, 0, stream>>>(ei, logits, lmax2, den2, E, Etot);
    edge_aggregate_kernel<1, 64><<<blks(T2), 256, 0, stream>>>(xl2, ei, logits, den2, out, E, Etot);
    bias_act_kernel<<<blks(N * 64), 256, 0, stream>>>(out, bias2, N * 64, 63, 0);
}